// Model_22265110462509
// MI455X (gfx1250) — compile-verified
//
#include <hip/hip_runtime.h>

// Problem constants (from the reference):
//   N=32, C=64, H=W=128, G=32, CG=2, HXW=16384, SWISH_SCALE=1
#define N_TOT   32
#define C_TOT   64
#define G_TOT   32
#define HXW     16384
#define HW4     (HXW / 4)            // 4096 float4 per channel plane
#define PLANE   (C_TOT * HXW)        // 1,048,576 elements per batch image
#define DX_ELEMS (N_TOT * PLANE)     // 33,554,432
#define DG_ELEMS (C_TOT * HXW)       // 1,048,576

// Native clang vector so __builtin_nontemporal_load/store accept it
// (HIP's float4 is a class type and is rejected by the builtin).
typedef float v4f __attribute__((ext_vector_type(4)));

__global__ __launch_bounds__(256)
void gn_swish_bwd_kernel(const float* __restrict__ dy,
                         const float* __restrict__ mean,
                         const float* __restrict__ rstd,
                         const float* __restrict__ x,
                         const float* __restrict__ gamma,
                         const float* __restrict__ beta,
                         float* __restrict__ out)
{
    // One thread owns 4 contiguous pixels of one channel across the whole batch.
    const int t   = blockIdx.x * 256 + threadIdx.x;  // 0 .. 262143
    const int c   = t >> 12;                         // t / 4096 (wave-uniform)
    const int hw4 = t & (HW4 - 1);
    const int base = c * HXW + (hw4 << 2);           // fits in 32-bit

    // g is identical across the wave: make it provably uniform so every
    // mean/rstd/gamma/beta access goes down the scalar (SMEM/KMcnt) path
    // instead of occupying vector-memory issue slots next to the b128 stream.
    const int gu = __builtin_amdgcn_readfirstlane(c >> 1);  // group index
    const int cu = __builtin_amdgcn_readfirstlane(c);

    const float gam = gamma[cu];
    const float bet = beta[cu];

    float* __restrict__ dx_out = out;
    float* __restrict__ dg_out = out + DX_ELEMS;
    float* __restrict__ db_out = out + DX_ELEMS + DG_ELEMS;

    v4f dg = (v4f)(0.0f);
    v4f db = (v4f)(0.0f);

    #pragma unroll 4
    for (int n = 0; n < N_TOT; ++n) {
        const int idx = n * PLANE + base;

        // The n-stride is 4 MB: explicitly prefetch 2 batches ahead
        // (lowers to global_prefetch_b8 on gfx1250).
        if (n + 2 < N_TOT) {
            __builtin_prefetch(dy + idx + 2 * PLANE, 0, 1);
            __builtin_prefetch(x  + idx + 2 * PLANE, 0, 1);
        }

        // Wave-uniform statistics -> scalar loads (s_load_b32).
        const float mu = mean[n * G_TOT + gu];
        const float rs = rstd[n * G_TOT + gu];

        // Streaming, single-use data: non-temporal b128 traffic so we do not
        // evict useful lines from the 192 MB L2 (working set is 392 MB).
        const v4f dy4 = __builtin_nontemporal_load((const v4f*)(dy + idx));
        const v4f x4  = __builtin_nontemporal_load((const v4f*)(x  + idx));

        v4f dx4;
        #pragma unroll
        for (int k = 0; k < 4; ++k) {
            const float xv   = x4[k];
            const float dyv  = dy4[k];
            const float xn   = (xv - mu) * rs;                  // x_norm
            const float o    = fmaf(xn, gam, bet);              // affine out
            const float e    = __expf(-o);                      // v_exp_f32 path
            const float sig  = __builtin_amdgcn_rcpf(1.0f + e); // v_rcp_f32
            const float sw   = o * sig;                         // swish
            const float grad = fmaf(sw, (1.0f - sig), sig);     // sig + swish*(1-sig)
            const float dys  = dyv * grad;                      // dy_swish
            dx4[k] = dys * gam * rs;                            // dx
            dg[k] += dys * xn;                                  // dgamma partial
            db[k] += dys;                                       // dbeta  partial
        }

        __builtin_nontemporal_store(dx4, (v4f*)(dx_out + idx));
    }

    // Reduction over n is complete in-registers: one clean b128 store each,
    // no atomics anywhere.
    __builtin_nontemporal_store(dg, (v4f*)(dg_out + base));
    __builtin_nontemporal_store(db, (v4f*)(db_out + base));
}

extern "C" void kernel_launch(void* const* d_in, const int* in_sizes, int n_in,
                              void* d_out, int out_size, void* d_ws, size_t ws_size,
                              hipStream_t stream)
{
    (void)in_sizes; (void)n_in; (void)d_ws; (void)ws_size; (void)out_size;

    const float* dy    = (const float*)d_in[0];
    const float* mean  = (const float*)d_in[1];
    const float* rstd  = (const float*)d_in[2];
    const float* x     = (const float*)d_in[3];
    const float* gamma = (const float*)d_in[4];
    const float* beta  = (const float*)d_in[5];
    float* out = (float*)d_out;

    // 262,144 threads total: 64 channels x 4096 float4-pixels.
    const int threads = 256;                    // 8 wave32 waves per block
    const int blocks  = (C_TOT * HW4) / threads; // 1024
    gn_swish_bwd_kernel<<<blocks, threads, 0, stream>>>(dy, mean, rstd, x,
                                                        gamma, beta, out);
}